// FeatureRouter_36275293782558
// MI455X (gfx1250) — compile-verified
//
#include <hip/hip_runtime.h>

#define HIDDEN 4096
#define LATENT 32768
#define TOKENS 4096
#define TOPK   64

typedef float v2f __attribute__((ext_vector_type(2)));
typedef float v4f __attribute__((ext_vector_type(4)));
typedef float v8f __attribute__((ext_vector_type(8)));

// ---------------------------------------------------------------------------
// Kernel 1: q[j] = sum_k W_q[j,k] * qv[k]   (one wave per row, coalesced k)
// ---------------------------------------------------------------------------
__global__ __launch_bounds__(256) void qvec_kernel(
    const float* __restrict__ Wq, const float* __restrict__ qv,
    float* __restrict__ q) {
  int wave = (blockIdx.x * blockDim.x + threadIdx.x) >> 5;
  int lane = threadIdx.x & 31;
  const float* row = Wq + (size_t)wave * HIDDEN;
  float s = 0.f;
  for (int k = lane; k < HIDDEN; k += 32) s = fmaf(row[k], qv[k], s);
  for (int off = 16; off; off >>= 1) s += __shfl_xor(s, off, 32);
  if (lane == 0) q[wave] = s;
}

// ---------------------------------------------------------------------------
// Kernel 2: scores[l] = sum_h q[h] * Dw[h,l]  via V_WMMA_F32_16X16X4_F32.
// A(16x4): q[k0..k0+3] replicated over M. B(4x16): Dw tile (rows k0..k0+3,
// cols l0..l0+15). D rows all equal; row 0 (VGPR0, lanes 0-15) -> scores.
// One wave per 16-column tile; EXEC is all-ones at every WMMA.
// ---------------------------------------------------------------------------
__global__ __launch_bounds__(256) void scores_wmma_kernel(
    const float* __restrict__ q, const float* __restrict__ Dw,
    float* __restrict__ scores) {
  int lane = threadIdx.x & 31;
  int tile = blockIdx.x * (blockDim.x >> 5) + (threadIdx.x >> 5);  // 0..2047
  int l0   = tile * 16;
  int half = lane >> 4;        // 0: K=0,1   1: K=2,3
  int n    = lane & 15;        // column within tile
  v8f c = {};
  for (int k0 = 0; k0 < HIDDEN; k0 += 4) {
    int kb = k0 + half * 2;
    v2f a, b;
    a.x = q[kb + 0];
    a.y = q[kb + 1];
    const float* bp = Dw + (size_t)kb * LATENT + (l0 + n);
    b.x = bp[0];
    b.y = bp[LATENT];
    c = __builtin_amdgcn_wmma_f32_16x16x4_f32(false, a, false, b,
                                              (short)0, c, false, false);
  }
  if (half == 0) scores[l0 + n] = c[0];   // D[M=0][N=n]
}

// ---------------------------------------------------------------------------
// Kernel 3: active[l] = any_t(z[t,l] > 0); apply -1e9 mask in place and init
// alpha to 1. Per-lane early exit: expected ~6 rows touched instead of 4096.
// ---------------------------------------------------------------------------
__global__ __launch_bounds__(256) void mask_alpha_kernel(
    const float* __restrict__ z, float* __restrict__ scores,
    float* __restrict__ alpha) {
  int l = blockIdx.x * blockDim.x + threadIdx.x;
  bool active = false;
  for (int t = 0; t < TOKENS; ++t) {
    if (__builtin_nontemporal_load(z + (size_t)t * LATENT + l) > 0.f) {
      active = true;
      break;
    }
  }
  if (!active) scores[l] = -1e9f;
  alpha[l] = 1.0f;
}

// ---------------------------------------------------------------------------
// Kernel 4: top-64 of 32768 scores -> alpha[idx] = 1 + 2*sigmoid(score*scale).
// Single 1024-thread workgroup; each thread owns 32 register-resident values;
// 64 rounds of (local argmax -> LDS tree argmax -> mark consumed).
// Tie-break on lowest index to match lax.top_k.
// ---------------------------------------------------------------------------
__global__ __launch_bounds__(1024) void topk_alpha_kernel(
    const float* __restrict__ scores, const float* __restrict__ log_scale,
    float* __restrict__ alpha) {
  __shared__ float sval[1024];
  __shared__ int   sidx[1024];
  int tid = threadIdx.x;
  float v[32];
  unsigned consumed = 0u;
#pragma unroll
  for (int i = 0; i < 32; ++i) v[i] = scores[tid + (i << 10)];
  float scale = fminf(expf(log_scale[0]), 10.0f);

  for (int r = 0; r < TOPK; ++r) {
    float bv = -INFINITY;
    int   bi = LATENT;
#pragma unroll
    for (int i = 0; i < 32; ++i) {
      if (!((consumed >> i) & 1u)) {
        float x = v[i];
        int  ix = tid + (i << 10);
        if (x > bv || (x == bv && ix < bi)) { bv = x; bi = ix; }
      }
    }
    sval[tid] = bv;
    sidx[tid] = bi;
    __syncthreads();
    for (int s = 512; s > 0; s >>= 1) {
      if (tid < s) {
        float ov = sval[tid + s];
        int   oi = sidx[tid + s];
        if (ov > sval[tid] || (ov == sval[tid] && oi < sidx[tid])) {
          sval[tid] = ov;
          sidx[tid] = oi;
        }
      }
      __syncthreads();
    }
    int   widx = sidx[0];
    float wval = sval[0];
    if (tid == 0) {
      float boost = 1.0f + 2.0f / (1.0f + expf(-wval * scale));
      alpha[widx] = boost;
    }
    if ((widx & 1023) == tid) consumed |= 1u << (widx >> 10);
    __syncthreads();
  }
}

// ---------------------------------------------------------------------------
// Kernel 5: out[t,l] = z[t,l] > 0 ? alpha[l] : 1. float4 streams, NT hints on
// the 1 GB of dead traffic; alpha stays regular-temporal (L2/WGP$ resident).
// ---------------------------------------------------------------------------
__global__ __launch_bounds__(256) void output_kernel(
    const v4f* __restrict__ z4, const v4f* __restrict__ alpha4,
    v4f* __restrict__ out4) {
  size_t i = (size_t)blockIdx.x * blockDim.x + threadIdx.x;
  v4f zz = __builtin_nontemporal_load(z4 + i);
  v4f aa = alpha4[i & (LATENT / 4 - 1)];
  v4f r;
  r.x = zz.x > 0.f ? aa.x : 1.0f;
  r.y = zz.y > 0.f ? aa.y : 1.0f;
  r.z = zz.z > 0.f ? aa.z : 1.0f;
  r.w = zz.w > 0.f ? aa.w : 1.0f;
  __builtin_nontemporal_store(r, out4 + i);
}

// ---------------------------------------------------------------------------
extern "C" void kernel_launch(void* const* d_in, const int* in_sizes, int n_in,
                              void* d_out, int out_size, void* d_ws,
                              size_t ws_size, hipStream_t stream) {
  (void)in_sizes; (void)n_in; (void)out_size; (void)ws_size;
  const float* qv = (const float*)d_in[0];
  const float* z  = (const float*)d_in[1];
  const float* Dw = (const float*)d_in[2];
  const float* Wq = (const float*)d_in[3];
  const float* ls = (const float*)d_in[4];
  float* out = (float*)d_out;

  float* q      = (float*)d_ws;       // HIDDEN floats
  float* scores = q + HIDDEN;         // LATENT floats
  float* alpha  = scores + LATENT;    // LATENT floats

  qvec_kernel<<<HIDDEN / 8, 256, 0, stream>>>(Wq, qv, q);
  scores_wmma_kernel<<<(LATENT / 16) / 8, 256, 0, stream>>>(q, Dw, scores);
  mask_alpha_kernel<<<LATENT / 256, 256, 0, stream>>>(z, scores, alpha);
  topk_alpha_kernel<<<1, 1024, 0, stream>>>(scores, ls, alpha);
  output_kernel<<<(size_t)TOKENS * (LATENT / 4) / 256, 256, 0, stream>>>(
      (const v4f*)z, (const v4f*)alpha, (v4f*)out);
}